// RoIPool3d_40407052320967
// MI455X (gfx1250) — compile-verified
//
#include <hip/hip_runtime.h>
#include <stdint.h>

// RoI 3D max-pool for MI455X (gfx1250, wave32).
// x: [N=2, C=256, D=16, H=64, W=64] f32, rois: [K,7] f32 -> out: [K,256,4,7,7] f32.
// Memory-bound; the 134MB input fits in the 192MB L2. Uses CDNA5 async
// global->LDS B128 DMA (ASYNCcnt) to stage each RoI sub-box per channel,
// then a wave32 shuffle-tree for the 7 width-bin maxima.

#define C_    256
#define D_    16
#define H_    64
#define W_    64
#define PD    4
#define PH    7
#define PW    7
#define NWAVE 8
#define CAP   1024   // floats of LDS staging per wave (worst-case padded box ~700)

__device__ __forceinline__ int iclamp(int v, int lo, int hi) {
  return v < lo ? lo : (v > hi ? hi : v);
}

// torchvision roi_pool bin bounds (float32 math, matching the JAX reference)
__device__ __forceinline__ void axis_bin(int start, int end, int pooled, int size,
                                         int p, int& lo, int& hi) {
  float len = fmaxf((float)(end - start + 1), 1.0f);
  float b   = len / (float)pooled;
  int l = (int)floorf((float)p * b) + start;
  int h = (int)ceilf(((float)p + 1.0f) * b) + start;
  lo = iclamp(l, 0, size);
  hi = iclamp(h, 0, size);
}

__global__ __launch_bounds__(256) void
roipool3d_kernel(const float* __restrict__ x, const float* __restrict__ rois,
                 float* __restrict__ out) {
  __shared__ float lbuf[NWAVE * CAP];

  const int bid  = blockIdx.x;
  const int k    = bid / (PD * PH);
  const int pd   = (bid / PH) % PD;
  const int ph   = bid % PH;
  const int tid  = threadIdx.x;
  const int wave = tid >> 5;
  const int lane = tid & 31;

  const float* r = rois + (size_t)k * 7;
  const int bi = (int)r[0];
  // round-to-nearest-even like jnp.round (rintf under default RNE mode)
  const int z1  = (int)rintf(r[1] * 0.125f);
  const int xx1 = (int)rintf(r[2] * 0.125f);
  const int y1  = (int)rintf(r[3] * 0.125f);
  const int z2  = (int)rintf(r[4] * 0.125f);
  const int xx2 = (int)rintf(r[5] * 0.125f);
  const int y2  = (int)rintf(r[6] * 0.125f);

  int dlo, dhi, hlo, hhi;
  axis_bin(z1, z2, PD, D_, pd, dlo, dhi);   // depth bin for this block
  axis_bin(y1, y2, PH, H_, ph, hlo, hhi);   // height bin for this block

  // union of all 7 width bins = [lo(0), hi(PW-1))
  int w0, w1, wtmp;
  axis_bin(xx1, xx2, PW, W_, 0,      w0,   wtmp);
  axis_bin(xx1, xx2, PW, W_, PW - 1, wtmp, w1);

  // per-lane width bin: lanes split as 4 row-groups x 8 w-bins (7 used)
  const int wb = lane & 7;
  const int rg = lane >> 3;
  int wlo_b, whi_b;
  axis_bin(xx1, xx2, PW, W_, (wb < PW) ? wb : 0, wlo_b, whi_b);

  const int dsz  = dhi - dlo;
  const int hsz  = hhi - hlo;
  const int rows = (dsz > 0 && hsz > 0) ? dsz * hsz : 0;

  // 16B-align the staged window: stays inside the W=64 row, so always in-bounds
  const int w0a = w0 & ~3;
  const int WLp = (w1 > w0a) ? ((w1 - w0a + 3) & ~3) : 0;  // padded row pitch (floats)
  const int cpr = WLp >> 2;                                 // 16B chunks per row
  const int E16 = rows * cpr;                               // total 16B chunks
  const bool staged = (E16 > 0) && (rows * WLp <= CAP);

  const unsigned ldsbase =
      (unsigned)(uintptr_t)(&lbuf[wave * CAP]);   // low 32 bits of flat = LDS offset

  for (int c = wave; c < C_; c += NWAVE) {
    const float* base = x + ((size_t)bi * C_ + c) * (size_t)(D_ * H_ * W_);

    // previous iteration's LDS reads must retire before the async engine rewrites LDS
    asm volatile("s_wait_dscnt 0" ::: "memory");

    if (staged) {
      for (int e = lane; e < E16; e += 32) {
        const int rr = e / cpr;            // row index within (d,h) box
        const int cc = e - rr * cpr;       // 16B chunk within row
        const int d  = dlo + rr / hsz;
        const int h  = hlo + (rr - (rr / hsz) * hsz);
        const unsigned long long ga = (unsigned long long)(uintptr_t)
            (base + ((size_t)(d * H_ + h) * W_ + (w0a + (cc << 2))));
        const unsigned la = ldsbase + (unsigned)e * 16u;
        // CDNA5 async DMA global -> LDS, 16B per lane, tracked on ASYNCcnt
        asm volatile("global_load_async_to_lds_b128 %0, %1, off"
                     :: "v"(la), "v"(ga) : "memory");
      }
    }
    asm volatile("s_wait_asynccnt 0" ::: "memory");

    float m = -__builtin_inff();
    const bool active = (wb < PW) && (rows > 0) && (whi_b > wlo_b);
    if (active) {
      if (staged) {
        const float* lb = &lbuf[wave * CAP];
        for (int rr2 = rg; rr2 < rows; rr2 += 4) {
          const float* rp = lb + rr2 * WLp - w0a;
          for (int w = wlo_b; w < whi_b; ++w) m = fmaxf(m, rp[w]);
        }
      } else {  // oversized box fallback: direct L2 reads
        for (int rr2 = rg; rr2 < rows; rr2 += 4) {
          const int d = dlo + rr2 / hsz;
          const int h = hlo + (rr2 - (rr2 / hsz) * hsz);
          const float* rp = base + (size_t)(d * H_ + h) * W_;
          for (int w = wlo_b; w < whi_b; ++w) m = fmaxf(m, rp[w]);
        }
      }
    }
    // fold the 4 row-groups (wave32 lane-xor shuffle tree)
    m = fmaxf(m, __shfl_xor(m, 8, 32));
    m = fmaxf(m, __shfl_xor(m, 16, 32));

    if (rg == 0 && wb < PW) {
      const bool empty = (dsz <= 0) || (hsz <= 0) || (whi_b <= wlo_b);
      out[(((size_t)k * C_ + c) * PD + pd) * (PH * PW) + ph * PW + wb] =
          empty ? 0.0f : m;
    }
  }
}

extern "C" void kernel_launch(void* const* d_in, const int* in_sizes, int n_in,
                              void* d_out, int out_size, void* d_ws, size_t ws_size,
                              hipStream_t stream) {
  (void)n_in; (void)d_ws; (void)ws_size; (void)out_size;
  const float* x    = (const float*)d_in[0];
  const float* rois = (const float*)d_in[1];
  float* out        = (float*)d_out;
  const int K = in_sizes[1] / 7;
  dim3 grid(K * PD * PH);   // 448 blocks for K=16
  roipool3d_kernel<<<grid, 256, 0, stream>>>(x, rois, out);
}